// Model_45380624450103
// MI455X (gfx1250) — compile-verified
//
#include <hip/hip_runtime.h>
#include <math.h>

typedef float v2f __attribute__((ext_vector_type(2)));
typedef float v8f __attribute__((ext_vector_type(8)));

#define BATCH 128
#define TSTEPS 512
#define NCH 64
#define IIN 4
#define DFEAT 256   // N*I
#define HID 512
#define GATES 2048  // 4*H
#define NOUT 256    // N*O

#define KC 128          // K-chunk staged in LDS
#define WROWS 64        // 4 gates x 16 cols staged per block
#define WPAD 132        // row stride in LDS floats: 132*4B = 528B (16B aligned, bank-safe)

__device__ __forceinline__ float sig_(float x) { return 1.0f / (1.0f + __expf(-x)); }

__device__ __forceinline__ v8f wmma_f32(v2f a, v2f b, v8f c) {
    // V_WMMA_F32_16X16X4_F32 : D = A(16x4 f32) * B(4x16 f32) + C(16x16 f32)
    return __builtin_amdgcn_wmma_f32_16x16x4_f32(
        /*neg_a=*/false, a, /*neg_b=*/false, b,
        /*c_mod=*/(short)0, c, /*reuse_a=*/false, /*reuse_b=*/false);
}

// ---------------------------------------------------------------------------
// Fused LSTM cell step (LDS-staged weights):
//   gates = A1(BxK1,ldA1) @ W1(GATESxK1)^T  [+ A2(BxK2) @ W2(GATESxK2)^T]
//           [+ base(BxGATES)] [+ bih + bhh]
//   c_new = sigmoid(f)*c_ref + sigmoid(i)*tanh(g);  h = sigmoid(o)*tanh(c_new)
// grid.x = HID/16 = 32 column blocks; block = 256 (8 waves); wave = m-tile.
// The 64 weight rows (4 gate sections x 16 cols) are shared by all 8 waves ->
// staged cooperatively through LDS in double-buffered K-chunks of 128.
// ---------------------------------------------------------------------------
__global__ __launch_bounds__(256)
void lstm_cell_fused(const float* __restrict__ A1, int ldA1, int K1,
                     const float* __restrict__ W1,
                     const float* __restrict__ A2, int K2,
                     const float* __restrict__ W2,
                     const float* __restrict__ base,
                     const float* __restrict__ bih, const float* __restrict__ bhh,
                     const float* __restrict__ c_ref,
                     float* __restrict__ h_out,
                     float* __restrict__ c_out) {
    const int j     = blockIdx.x;            // column block within H
    const int wave  = threadIdx.x >> 5;      // m-tile 0..7
    const int lane  = threadIdx.x & 31;
    const int lrow  = lane & 15;
    const int hi    = lane >> 4;             // 0: K0-1 / M0-7, 1: K2-3 / M8-15
    const int khalf = hi * 2;
    const int m0    = wave * 16;
    const int col0  = j * 16;
    const int col   = col0 + lrow;

    __shared__ float sW[2][WROWS * WPAD];    // ~68 KB of the 320 KB WGP LDS

    v8f acc[4];
    #pragma unroll
    for (int s = 0; s < 4; ++s) {
        float bv = 0.0f;
        int g = s * HID + col;
        if (bih) bv += bih[g];
        if (bhh) bv += bhh[g];
        #pragma unroll
        for (int r = 0; r < 8; ++r) acc[s][r] = bv;
    }

    // Coalesced cooperative copy of one K-chunk of the 64 shared weight rows.
    auto stage = [&](const float* __restrict__ Wsrc, int K, int k0c, int buf) {
        #pragma unroll
        for (int it = 0; it < (WROWS * KC / 4) / 256; ++it) {   // 8 float4 / thread
            int idx  = threadIdx.x + 256 * it;                  // 0..2047
            int row  = idx >> 5;                                // KC/4 = 32 quads/row
            int kq   = (idx & 31) << 2;
            int rowg = (row >> 4) * HID + col0 + (row & 15);    // gate s = row>>4
            float4 v = *(const float4*)(Wsrc + (size_t)rowg * K + k0c + kq);
            *(float4*)&sW[buf][row * WPAD + kq] = v;
        }
    };

    // K-loop over one (A, W) operand pair with double-buffered LDS staging.
    auto runK = [&](const float* __restrict__ A, int ldA,
                    const float* __restrict__ W, int K) {
        const int nch = K / KC;
        stage(W, K, 0, 0);
        for (int c = 0; c < nch; ++c) {
            __syncthreads();                         // chunk c staged; buf (c+1)&1 free of readers
            if (c + 1 < nch) stage(W, K, (c + 1) * KC, (c + 1) & 1);
            const float* sb = &sW[c & 1][0];
            const int k0c = c * KC;
            for (int kk = 0; kk < KC; kk += 4) {
                v2f a = *(const v2f*)(A + (size_t)(m0 + lrow) * ldA + k0c + kk + khalf);
                #pragma unroll
                for (int s = 0; s < 4; ++s) {
                    v2f b = *(const v2f*)(sb + (s * 16 + lrow) * WPAD + kk + khalf);
                    acc[s] = wmma_f32(a, b, acc[s]);
                }
            }
        }
        __syncthreads();                             // protect buf0 before next runK restages
    };

    runK(A1, ldA1, W1, K1);
    if (K2 > 0) runK(A2, K2, W2, K2);

    #pragma unroll
    for (int r = 0; r < 8; ++r) {
        int m = m0 + r + hi * 8;
        float gi = acc[0][r], gf = acc[1][r], gg = acc[2][r], go = acc[3][r];
        if (base) {
            const float* bp = base + (size_t)m * GATES;
            gi += bp[0 * HID + col];
            gf += bp[1 * HID + col];
            gg += bp[2 * HID + col];
            go += bp[3 * HID + col];
        }
        float i_ = sig_(gi), f_ = sig_(gf), o_ = sig_(go);
        float g_ = tanhf(gg);
        float cn = f_ * c_ref[(size_t)m * HID + col] + i_ * g_;
        h_out[(size_t)m * HID + col] = o_ * tanhf(cn);
        if (c_out) c_out[(size_t)m * HID + col] = cn;
    }
}

// ---------------------------------------------------------------------------
// Plain GEMM + bias: C(128 x Ncols, ldC) = A(128 x K, ldA) @ W(Ncols x K)^T + b
// grid.x = Ncols/16, block 256 (8 waves = 8 m-tiles)
// ---------------------------------------------------------------------------
__global__ __launch_bounds__(256)
void gemm_bias(const float* __restrict__ A, int ldA, int K,
               const float* __restrict__ W,
               const float* __restrict__ bias, const float* __restrict__ bias2,
               float* __restrict__ C, int ldC) {
    const int n0    = blockIdx.x * 16;
    const int wave  = threadIdx.x >> 5;
    const int lane  = threadIdx.x & 31;
    const int lrow  = lane & 15;
    const int hi    = lane >> 4;
    const int khalf = hi * 2;
    const int m0    = wave * 16;

    float bv = 0.0f;
    if (bias)  bv += bias[n0 + lrow];
    if (bias2) bv += bias2[n0 + lrow];
    v8f acc;
    #pragma unroll
    for (int r = 0; r < 8; ++r) acc[r] = bv;

    for (int k0 = 0; k0 < K; k0 += 4) {
        v2f a = *(const v2f*)(A + (size_t)(m0 + lrow) * ldA + k0 + khalf);
        v2f b = *(const v2f*)(W + (size_t)(n0 + lrow) * K + k0 + khalf);
        acc = wmma_f32(a, b, acc);
    }
    #pragma unroll
    for (int r = 0; r < 8; ++r) {
        int m = m0 + r + hi * 8;
        C[(size_t)m * ldC + n0 + lrow] = acc[r];
    }
}

// (B,T,N,I) -> X (T,B,D) with D index d = i*N + n  (permute (0,1,3,2) + flatten)
__global__ void permute_kernel(const float* __restrict__ in, float* __restrict__ X) {
    size_t idx = (size_t)blockIdx.x * blockDim.x + threadIdx.x;
    if (idx >= (size_t)BATCH * TSTEPS * DFEAT) return;
    int d = idx % DFEAT;
    size_t bt = idx / DFEAT;
    int b = bt % BATCH;
    int t = bt / BATCH;          // X is (T,B,D): idx = (t*B + b)*D + d
    int i = d / NCH, n = d % NCH;
    X[idx] = in[(((size_t)b * TSTEPS + t) * NCH + n) * IIN + i];
}

__global__ void zero_kernel(float* __restrict__ p, int n) {
    int i = blockIdx.x * blockDim.x + threadIdx.x;
    if (i < n) p[i] = 0.0f;
}

__global__ void relu_kernel(const float* __restrict__ a, float* __restrict__ o, int n) {
    int i = blockIdx.x * blockDim.x + threadIdx.x;
    if (i < n) o[i] = fmaxf(a[i], 0.0f);
}

extern "C" void kernel_launch(void* const* d_in, const int* in_sizes, int n_in,
                              void* d_out, int out_size, void* d_ws, size_t ws_size,
                              hipStream_t stream) {
    (void)in_sizes; (void)n_in; (void)out_size; (void)ws_size;
    const float* inputs   = (const float*)d_in[0];
    const float* eWih0    = (const float*)d_in[1];
    const float* eWhh0    = (const float*)d_in[2];
    const float* ebih0    = (const float*)d_in[3];
    const float* ebhh0    = (const float*)d_in[4];
    const float* eWih1    = (const float*)d_in[5];
    const float* eWhh1    = (const float*)d_in[6];
    const float* ebih1    = (const float*)d_in[7];
    const float* ebhh1    = (const float*)d_in[8];
    const float* dinWih   = (const float*)d_in[9];
    const float* dinWhh   = (const float*)d_in[10];
    const float* dinbih   = (const float*)d_in[11];
    const float* dinbhh   = (const float*)d_in[12];
    const float* doutWih  = (const float*)d_in[13];
    const float* doutWhh  = (const float*)d_in[14];
    const float* doutbih  = (const float*)d_in[15];
    const float* doutbhh  = (const float*)d_in[16];
    const float* projW    = (const float*)d_in[17];
    const float* projb    = (const float*)d_in[18];
    float* out = (float*)d_out;

    const size_t BH = (size_t)BATCH * HID;           // 65536
    const size_t BG = (size_t)BATCH * GATES;         // 262144
    float* ws   = (float*)d_ws;
    float* X    = ws;                                 // T*B*D
    float* zbuf = X + (size_t)TSTEPS * BATCH * DFEAT; // B*H zeros
    float* h0b  = zbuf + BH;                          // 2 * B*H (ping-pong)
    float* c0b  = h0b + 2 * BH;
    float* h1b  = c0b + 2 * BH;
    float* c1b  = h1b + 2 * BH;
    float* gin  = c1b + 2 * BH;                       // B*GATES decoder base (layer din)
    float* gout = gin + BG;                           // B*GATES decoder base (layer dout)
    float* hd1  = gout + BG;                          // B*H
    float* hd2  = hd1 + BH;                           // B*H

    // zero scratch used as h_{-1}/c_{-1}/p_{-1}
    zero_kernel<<<(int)((BH + 255) / 256), 256, 0, stream>>>(zbuf, (int)BH);

    // permute inputs -> X (T,B,D)
    {
        size_t tot = (size_t)BATCH * TSTEPS * DFEAT;
        permute_kernel<<<(int)((tot + 255) / 256), 256, 0, stream>>>(inputs, X);
    }

    dim3 blk(256);
    dim3 grdCell(HID / 16);      // 32

    // ---- encoder: interleave layer0/layer1 per step (no full hs0 storage) ----
    for (int t = 0; t < TSTEPS; ++t) {
        const float* h0p = t ? (h0b + ((t - 1) & 1) * BH) : zbuf;
        const float* c0p = t ? (c0b + ((t - 1) & 1) * BH) : zbuf;
        const float* h1p = t ? (h1b + ((t - 1) & 1) * BH) : zbuf;
        const float* c1p = t ? (c1b + ((t - 1) & 1) * BH) : zbuf;
        float* h0n = h0b + (t & 1) * BH;
        float* c0n = c0b + (t & 1) * BH;
        float* h1n = h1b + (t & 1) * BH;
        float* c1n = c1b + (t & 1) * BH;

        lstm_cell_fused<<<grdCell, blk, 0, stream>>>(
            X + (size_t)t * BATCH * DFEAT, DFEAT, DFEAT, eWih0,
            h0p, HID, eWhh0, nullptr, ebih0, ebhh0, c0p, h0n, c0n);
        lstm_cell_fused<<<grdCell, blk, 0, stream>>>(
            h0n, HID, HID, eWih1,
            h1p, HID, eWhh1, nullptr, ebih1, ebhh1, c1p, h1n, c1n);
    }
    const float* hT0 = h0b + BH;   // parity of t=511
    const float* cT0 = c0b + BH;
    const float* hT1 = h1b + BH;
    const float* cT1 = c1b + BH;

    // ---- decoder: recurrent-state gate contributions are constant -> hoist ----
    gemm_bias<<<dim3(GATES / 16), blk, 0, stream>>>(
        hT0, HID, HID, dinWhh, dinbih, dinbhh, gin, GATES);
    gemm_bias<<<dim3(GATES / 16), blk, 0, stream>>>(
        hT1, HID, HID, doutWhh, doutbih, doutbhh, gout, GATES);

    for (int t = 0; t < TSTEPS; ++t) {
        const float* pprev = t ? (out + (size_t)(t - 1) * NOUT) : zbuf;
        int ldp = t ? (TSTEPS * NOUT) : NOUT;
        // h1 = cell(p_prev | fixed hT0,cT0)
        lstm_cell_fused<<<grdCell, blk, 0, stream>>>(
            pprev, ldp, NOUT, dinWih,
            nullptr, 0, nullptr, gin, nullptr, nullptr, cT0, hd1, nullptr);
        // h2 = cell(h1 | fixed hT1,cT1)
        lstm_cell_fused<<<grdCell, blk, 0, stream>>>(
            hd1, HID, HID, doutWih,
            nullptr, 0, nullptr, gout, nullptr, nullptr, cT1, hd2, nullptr);
        // p_t = h2 @ projW^T + projb -> strided directly into outputs[b, t, :]
        gemm_bias<<<dim3(NOUT / 16), blk, 0, stream>>>(
            hd2, HID, HID, projW, projb, nullptr,
            out + (size_t)t * NOUT, TSTEPS * NOUT);
    }

    // embedding = relu(hT1) appended after outputs
    relu_kernel<<<(int)((BH + 255) / 256), 256, 0, stream>>>(
        hT1, out + (size_t)BATCH * TSTEPS * NOUT, (int)BH);
}